// NMRModel_11287174054046
// MI455X (gfx1250) — compile-verified
//
#include <hip/hip_runtime.h>

// ---------------------------------------------------------------------------
// NMR 2-spin propagation on MI455X, restructured as a 2-level parallel scan:
//   (1) k_expm:       all 65536 propagators U_i = exp(-i*dt*H_i) in parallel
//                     (9-term Horner-Taylor, ||dt*H|| ~ 0.07 -> fp32-exact)
//   (2) k_chunk_scan: 256-matrix Kogge-Stone scan per workgroup, entirely in
//                     LDS (2 x 32KB ping-pong buffers), products computed as
//                     V_WMMA_F32_16X16X4_F32 on the real 8x8 embedding
//                     [[X,-Y],[Y,X]], two 8x8x8 products per 16x16 WMMA tile.
//                     Writes local prefixes + chunk total.
//   (3) k_chunk_scan again (1 block) on the 256 chunk totals.
//   (4) k_apply_measure: P_i = local_i * chunkPrefix, fused with the
//                     measurement m_n = 0.25*tr(conj(Op) . P rho0 P^H)
//                     via a small LDS staging buffer.
// 4 kernel launches total (vs 18 for flat Kogge-Stone) -- the problem is
// launch-latency bound: 16MB working set lives in the 192MB L2, ~2 GFLOP of
// fp32 WMMA.  FP32 WMMA (not f16/bf16) is required: the product of 65536
// unitaries must remain unitary to fp32 precision.
//
// Storage per matrix: 32 floats = X (Re, col-major 4x4) then Y (Im, col-major).
// ---------------------------------------------------------------------------

#define NS 65536
#define CHUNK 256
#define DT 1e-5f

typedef float v2f __attribute__((ext_vector_type(2)));
typedef float v8f __attribute__((ext_vector_type(8)));

// ---------------------------------------------------------------------------
// Kernel 1: per-step propagator via 9-term Horner-Taylor expm
// ---------------------------------------------------------------------------
__global__ void k_expm(const float* __restrict__ ux, const float* __restrict__ uy,
                       const float* __restrict__ v1p, const float* __restrict__ v2p,
                       const float* __restrict__ Jp, float* __restrict__ P0) {
  const int n = blockIdx.x * blockDim.x + threadIdx.x;
  const float PI = 3.14159265358979f;
  const float v1 = v1p[0], v2 = v2p[0], J = Jp[0];

  const float d0 =  PI * (v1 + v2) + 0.5f * PI * J;
  const float d1 =  PI * (v1 - v2) - 0.5f * PI * J;
  const float d2 =  PI * (v2 - v1) - 0.5f * PI * J;
  const float d3 = -PI * (v1 + v2) + 0.5f * PI * J;
  const float jc =  PI * J;
  const float er =  0.5f * ux[n];               // e = 0.5*(ux - i*uy)
  const float ei = -0.5f * uy[n];

  float Hr[4][4] = {{d0, er, er, 0.f},
                    {er, d1, jc, er},
                    {er, jc, d2, er},
                    {0.f, er, er, d3}};
  float Hi[4][4] = {{0.f,  ei,  ei, 0.f},
                    {-ei, 0.f, 0.f,  ei},
                    {-ei, 0.f, 0.f,  ei},
                    {0.f, -ei, -ei, 0.f}};

  float Ar[4][4], Ai[4][4], Sr[4][4], Si[4][4];
#pragma unroll
  for (int r = 0; r < 4; ++r)
#pragma unroll
    for (int c = 0; c < 4; ++c) {
      Ar[r][c] = DT * Hi[r][c];                 // A = -i*dt*H
      Ai[r][c] = -DT * Hr[r][c];
      Sr[r][c] = (r == c) ? 1.f : 0.f;
      Si[r][c] = 0.f;
    }

#pragma unroll
  for (int k = 9; k >= 1; --k) {
    float Tr[4][4], Ti[4][4];
#pragma unroll
    for (int r = 0; r < 4; ++r)
#pragma unroll
      for (int c = 0; c < 4; ++c) {
        float tr = 0.f, ti = 0.f;
#pragma unroll
        for (int m = 0; m < 4; ++m) {
          tr = fmaf(Ar[r][m], Sr[m][c], fmaf(-Ai[r][m], Si[m][c], tr));
          ti = fmaf(Ar[r][m], Si[m][c], fmaf( Ai[r][m], Sr[m][c], ti));
        }
        Tr[r][c] = tr; Ti[r][c] = ti;
      }
    const float invk = 1.0f / (float)k;
#pragma unroll
    for (int r = 0; r < 4; ++r)
#pragma unroll
      for (int c = 0; c < 4; ++c) {
        Sr[r][c] = ((r == c) ? 1.f : 0.f) + Tr[r][c] * invk;
        Si[r][c] = Ti[r][c] * invk;
      }
  }

  float* o = P0 + (size_t)n * 32;
#pragma unroll
  for (int c = 0; c < 4; ++c)
#pragma unroll
    for (int r = 0; r < 4; ++r) {
      o[c * 4 + r]      = Sr[r][c];
      o[16 + c * 4 + r] = Si[r][c];
    }
}

// ---------------------------------------------------------------------------
// 8x8 real embedding element [[X,-Y],[Y,X]][r][c]; M = 32 floats (X, Y colmaj)
// Generic pointer: works for both global and LDS sources.
// ---------------------------------------------------------------------------
__device__ __forceinline__ float m8elem(const float* M, int r, int c) {
  const int rr = r & 3, cc = c & 3;
  const float x = M[cc * 4 + rr];
  const float y = M[16 + cc * 4 + rr];
  return (r < 4) ? ((c < 4) ? x : -y)
                 : ((c < 4) ? y :  x);
}

// ---------------------------------------------------------------------------
// Two packed 8x8x8 products (block-diagonal in one 16x16 tile) via two
// V_WMMA_F32_16X16X4_F32 (K = 4+4).  leftA/rightB are per-lane matrix
// pointers; useId substitutes the identity for the right operand (loads still
// issue from the clamped pointer -> no exec-mask branches around the loads).
// ---------------------------------------------------------------------------
__device__ __forceinline__ v8f mm_pair(const float* leftA, const float* rightB,
                                       bool useId, int L) {
  const int row16 = L & 15;
  const int rA  = row16 & 7;
  const int kkA = (L >= 16) ? 2 : 0;   // A: lanes 0-15 hold K0/K1, 16-31 K2/K3
  const int nB  = row16 & 7;
  const int kB  = (L >= 16) ? 1 : 0;   // B: v0 -> K0/K1, v1 -> K2/K3
  v8f acc = {};
#pragma unroll
  for (int h = 0; h < 2; ++h) {
    v2f a, b;
    a.x = m8elem(leftA, rA, 4 * h + kkA + 0);
    a.y = m8elem(leftA, rA, 4 * h + kkA + 1);
    const int k0 = 4 * h + 0 + kB;
    const int k1 = 4 * h + 2 + kB;
    const float b0 = m8elem(rightB, k0, nB);   // unconditional load
    const float b1 = m8elem(rightB, k1, nB);
    b.x = useId ? ((k0 == nB) ? 1.f : 0.f) : b0;
    b.y = useId ? ((k1 == nB) ? 1.f : 0.f) : b1;
    acc = __builtin_amdgcn_wmma_f32_16x16x4_f32(
        false, a, false, b, (short)0, acc, false, false);
  }
  return acc;
}

// Write the two diagonal 8x8 result blocks back in (X,Y) compact form.
// Product 0 cols 0-3 live in lanes 0-3; product 1 cols 0-3 in lanes 24-27.
__device__ __forceinline__ void store_pair(float* dst0, float* dst1,
                                           v8f acc, int L) {
  const bool active = (L < 4) || (L >= 24 && L < 28);
  if (active) {
    float* o = (L < 16) ? dst0 : dst1;
    const int cc = L & 7;                       // 0..3
    o[cc * 4 + 0] = acc[0];  o[cc * 4 + 1] = acc[1];
    o[cc * 4 + 2] = acc[2];  o[cc * 4 + 3] = acc[3];
    o[16 + cc * 4 + 0] = acc[4];  o[16 + cc * 4 + 1] = acc[5];
    o[16 + cc * 4 + 2] = acc[6];  o[16 + cc * 4 + 3] = acc[7];
  }
}

// ---------------------------------------------------------------------------
// Kernel 2/3: LDS-resident Kogge-Stone scan of CHUNK=256 matrices.
// 1024 threads = 32 waves; each step: 128 tiles (2 products each), 4 per wave.
// Dynamic LDS: 2 buffers x 256 x 128B = 64KB (WGP has 320KB).
// Writes the 256 inclusive local prefixes and the chunk total.
// ---------------------------------------------------------------------------
__global__ void __launch_bounds__(1024)
k_chunk_scan(const float* __restrict__ src, float* __restrict__ prefixes,
             float* __restrict__ totals) {
  extern __shared__ float smem[];
  float* buf0 = smem;
  float* buf1 = smem + CHUNK * 32;

  const int tid = threadIdx.x;
  const int wv  = tid >> 5;
  const int L   = tid & 31;
  const size_t base = (size_t)blockIdx.x * CHUNK * 32;

  // cooperative load: 8192 floats as float4s
  {
    const float4* gs = (const float4*)(src + base);
    float4* ls = (float4*)buf0;
    for (int j = tid; j < CHUNK * 8; j += 1024) ls[j] = gs[j];
  }
  __syncthreads();

  int cur = 0;
  for (int d = 1; d < CHUNK; d <<= 1) {         // 8 steps
    const float* S = cur ? buf1 : buf0;
    float*       D = cur ? buf0 : buf1;
#pragma unroll
    for (int j = 0; j < CHUNK / 64; ++j) {      // 4 tiles per wave
      const int t  = wv * (CHUNK / 64) + j;
      const int i0 = 2 * t;
      const int p  = (L & 15) >> 3;             // which product this lane feeds
      const int iA = i0 + p;
      const int ri = iA - d;
      const bool useId = (ri < 0);
      const float* leftA  = S + iA * 32;
      const float* rightB = S + (useId ? 0 : ri) * 32;
      v8f acc = mm_pair(leftA, rightB, useId, L);
      store_pair(D + i0 * 32, D + (i0 + 1) * 32, acc, L);
    }
    __syncthreads();
    cur ^= 1;
  }
  // 8 steps -> result back in buf0 (cur == 0)
  const float* R = cur ? buf1 : buf0;
  {
    const float4* ls = (const float4*)R;
    float4* gp = (float4*)(prefixes + base);
    for (int j = tid; j < CHUNK * 8; j += 1024) gp[j] = ls[j];
    if (tid < 8) {
      float4* gt = (float4*)(totals + (size_t)blockIdx.x * 32);
      gt[tid] = ((const float4*)(R + (CHUNK - 1) * 32))[tid];
    }
  }
}

// ---------------------------------------------------------------------------
// Kernel 4: P_i = local_i * G1[chunk-1] (identity for chunk 0), fused with
// the measurement.  Final fragments staged through 2KB LDS per block so
// lanes 0/1 of each wave read columns 0 and 3 of their matrices.
//   rho0 = diag(1,0,0,-1); conj(Op) nonzero (=1) at (0,1),(0,2),(1,3),(2,3)
//   rho[i][j] = P[i][0]conj(P[j][0]) - P[i][3]conj(P[j][3])
// ---------------------------------------------------------------------------
__global__ void k_apply_measure(const float* __restrict__ G0,
                                const float* __restrict__ G1,
                                float* __restrict__ out) {
  __shared__ float sc[8 * 64];                  // 8 waves x 2 matrices x 32
  const int tid = threadIdx.x;
  const int L   = tid & 31;
  const int wvb = tid >> 5;
  const int w   = (blockIdx.x * blockDim.x + tid) >> 5;  // global wave
  const int p   = (L & 15) >> 3;
  const int i   = 2 * w + p;
  const int c   = i >> 8;                        // chunk id (wave-uniform)
  const bool useId = (c == 0);
  const float* leftA  = G0 + (size_t)i * 32;
  const float* rightB = G1 + (size_t)(useId ? 0 : (c - 1)) * 32;

  v8f acc = mm_pair(leftA, rightB, useId, L);
  store_pair(sc + wvb * 64, sc + wvb * 64 + 32, acc, L);
  __syncthreads();

  if (L < 2) {
    const float* M = sc + wvb * 64 + L * 32;    // final P for matrix 2w+L
    float mr = 0.f, mi = 0.f;
    const int ii[4] = {0, 0, 1, 2};
    const int jj[4] = {1, 2, 3, 3};
#pragma unroll
    for (int t = 0; t < 4; ++t) {
      const int a = ii[t], b = jj[t];
      // column 0: X[r] = M[r], Y[r] = M[16+r];  column 3: M[12+r], M[28+r]
      mr += M[a] * M[b] + M[16 + a] * M[16 + b];
      mi += M[16 + a] * M[b] - M[a] * M[16 + b];
      mr -= M[12 + a] * M[12 + b] + M[28 + a] * M[28 + b];
      mi -= M[28 + a] * M[12 + b] - M[12 + a] * M[28 + b];
    }
    const int n = 2 * w + L;
    out[2 * n + 0] = 0.25f * mr;
    out[2 * n + 1] = 0.25f * mi;
  }
}

// ---------------------------------------------------------------------------
extern "C" void kernel_launch(void* const* d_in, const int* in_sizes, int n_in,
                              void* d_out, int out_size, void* d_ws, size_t ws_size,
                              hipStream_t stream) {
  const float* ux = (const float*)d_in[0];
  const float* uy = (const float*)d_in[1];
  const float* v1 = (const float*)d_in[2];
  const float* v2 = (const float*)d_in[3];
  const float* J  = (const float*)d_in[4];

  float* bufU = (float*)d_ws;                      // 8 MB: propagators
  float* G0   = bufU + (size_t)NS * 32;            // 8 MB: in-chunk prefixes
  float* T    = G0 + (size_t)NS * 32;              // 32 KB: chunk totals (256)
  float* G1   = T + (size_t)CHUNK * 32;            // 32 KB: scanned totals
  float* Td   = G1 + (size_t)CHUNK * 32;           // 128 B: dummy total

  const size_t lds_bytes = 2u * CHUNK * 32 * sizeof(float);  // 64 KB

  // (1) all propagators in parallel
  k_expm<<<NS / 256, 256, 0, stream>>>(ux, uy, v1, v2, J, bufU);

  // (2) 256 workgroups: LDS Kogge-Stone scan of each 256-matrix chunk
  k_chunk_scan<<<NS / CHUNK, 1024, lds_bytes, stream>>>(bufU, G0, T);

  // (3) one workgroup: scan the 256 chunk totals (same kernel)
  k_chunk_scan<<<1, 1024, lds_bytes, stream>>>(T, G1, Td);

  // (4) apply chunk prefixes + fused measurement -> complex64 output
  const int waves = NS / 2;
  k_apply_measure<<<(waves * 32) / 256, 256, 0, stream>>>(G0, G1, (float*)d_out);
}